// context2query_attention_6047313952906
// MI455X (gfx1250) — compile-verified
//
#include <hip/hip_runtime.h>

// Problem constants (match reference setup_inputs)
#define B_   32
#define LC   2048
#define LQ   256
#define DD   128
#define MASKV (-1e30f)

typedef __attribute__((ext_vector_type(2))) float v2f;
typedef __attribute__((ext_vector_type(8))) float v8f;

// D = A(16x4) * B(4x16) + C, full fp32 CDNA5 WMMA.
__device__ __forceinline__ v8f wmma_f32(v2f a, v2f b, v8f c) {
  return __builtin_amdgcn_wmma_f32_16x16x4_f32(false, a, false, b, (short)0, c,
                                               false, false);
}

__device__ __forceinline__ void atomicMaxF(float* a, float v) {
  if (v >= 0.0f) atomicMax((int*)a, __float_as_int(v));
  else           atomicMin((unsigned int*)a, __float_as_uint(v));
}

#define XCSTR 130   // 32/64-row x_cont tile stride (130 mod 64 == 2 -> b64 conflict-free)
#define PSTR  33    // kernel2 p-staging stride
#define PSTR3 258   // kernel3 S/p row stride (258 mod 64 == 2)

// ---------------------------------------------------------------------------
// k0: s0 = xc.W0 ; s1q = xq.W1 + bias ; init colmax/colsum/A (every call)
// ---------------------------------------------------------------------------
__global__ __launch_bounds__(256) void k0_init(
    const float* __restrict__ xc, const float* __restrict__ xq,
    const float* __restrict__ W0, const float* __restrict__ W1,
    const float* __restrict__ bias,
    float* __restrict__ s0, float* __restrict__ s1q,
    float* __restrict__ colmax, float* __restrict__ colsum,
    float* __restrict__ Aws) {
  int idx = blockIdx.x * 256 + threadIdx.x;
  if (idx < B_ * LC) {
    const float4* r4 = (const float4*)(xc + (size_t)idx * DD);
    const float4* w4 = (const float4*)W0;
    float acc = 0.f;
    for (int k = 0; k < DD / 4; ++k) {
      float4 a = r4[k], w = w4[k];
      acc += a.x * w.x + a.y * w.y + a.z * w.z + a.w * w.w;
    }
    s0[idx] = acc;
  }
  if (idx < B_ * LQ) {
    const float4* r4 = (const float4*)(xq + (size_t)idx * DD);
    const float4* w4 = (const float4*)W1;
    float acc = 0.f;
    for (int k = 0; k < DD / 4; ++k) {
      float4 a = r4[k], w = w4[k];
      acc += a.x * w.x + a.y * w.y + a.z * w.z + a.w * w.w;
    }
    s1q[idx]   = acc + bias[0];
    colmax[idx] = -3.0e38f;
    colsum[idx] = 0.f;
  }
  if (idx < B_ * LQ * DD) Aws[idx] = 0.f;
}

// ---------------------------------------------------------------------------
// k1: per-(b,q) column max of S + MASK*(1-c_mask)   (grid = 32 b * 32 ctiles)
// ---------------------------------------------------------------------------
__global__ __launch_bounds__(256) void k1_colmax(
    const float* __restrict__ xc, const float* __restrict__ xq,
    const int* __restrict__ cmask, const float* __restrict__ W2,
    const float* __restrict__ s0, const float* __restrict__ s1q,
    float* __restrict__ colmax) {
  __shared__ float xcw[64 * XCSTR];            // x_cont * W2
  __shared__ float s0l[64], cml[64], s1l[LQ];
  int b = blockIdx.x >> 5, ct = blockIdx.x & 31, c0 = ct * 64;
  int tid = threadIdx.x;
  for (int i = tid; i < 64 * DD; i += 256) {
    int r = i >> 7, c = i & 127;
    xcw[r * XCSTR + c] = xc[(size_t)(b * LC + c0 + r) * DD + c] * W2[c];
  }
  if (tid < 64) {
    s0l[tid] = s0[b * LC + c0 + tid];
    cml[tid] = cmask[b * LC + c0 + tid] ? 0.f : MASKV;
  }
  if (tid < LQ) s1l[tid] = s1q[b * LQ + tid];
  __syncthreads();

  int lane = tid & 31, wv = tid >> 5, half = lane >> 4, lr = lane & 15;
  const float* xqb = xq + (size_t)b * LQ * DD;
  for (int t = 0; t < 8; ++t) {
    int id = wv * 8 + t;
    int cs = (id & 3) * 16;        // c subtile
    int qs = (id >> 2) * 16;       // q subtile (0..240)
    v8f acc = {};
    for (int k0 = 0; k0 < DD; k0 += 4) {
      int kk = k0 + half * 2;
      v2f a = *(const v2f*)&xcw[(cs + lr) * XCSTR + kk];
      v2f bb = *(const v2f*)(xqb + (size_t)(qs + lr) * DD + kk);
      acc = wmma_f32(a, bb, acc);
    }
    int qg = qs + lr;
    float m = -3.0e38f;
    for (int r = 0; r < 8; ++r) {
      int cl = cs + half * 8 + r;
      m = fmaxf(m, acc[r] + s0l[cl] + cml[cl] + s1l[qg]);
    }
    m = fmaxf(m, __shfl_xor(m, 16, 32));       // combine row halves (wave32)
    if (lane < 16) atomicMaxF(&colmax[b * LQ + qg], m);
  }
}

// ---------------------------------------------------------------------------
// k2: colsum += sum_c exp(v - colmax); A[b,q,:] += p^T @ x_cont  (atomics)
// ---------------------------------------------------------------------------
__global__ __launch_bounds__(256) void k2_colacc(
    const float* __restrict__ xc, const float* __restrict__ xq,
    const int* __restrict__ cmask, const float* __restrict__ W2,
    const float* __restrict__ s0, const float* __restrict__ s1q,
    const float* __restrict__ colmax, float* __restrict__ colsum,
    float* __restrict__ Aws) {
  __shared__ float xcl[64 * XCSTR];            // raw x_cont
  __shared__ float w2l[DD];
  __shared__ float s0l[64], cml[64], s1l[LQ], cmxl[LQ];
  __shared__ float pst[64 * PSTR];             // p tile (64 c x 32 q)
  int b = blockIdx.x >> 5, ct = blockIdx.x & 31, c0 = ct * 64;
  int tid = threadIdx.x;
  for (int i = tid; i < 64 * DD; i += 256) {
    int r = i >> 7, c = i & 127;
    xcl[r * XCSTR + c] = xc[(size_t)(b * LC + c0 + r) * DD + c];
  }
  if (tid < DD) w2l[tid] = W2[tid];
  if (tid < 64) {
    s0l[tid] = s0[b * LC + c0 + tid];
    cml[tid] = cmask[b * LC + c0 + tid] ? 0.f : MASKV;
  }
  if (tid < LQ) { s1l[tid] = s1q[b * LQ + tid]; cmxl[tid] = colmax[b * LQ + tid]; }
  __syncthreads();

  int lane = tid & 31, wv = tid >> 5, half = lane >> 4, lr = lane & 15;
  const float* xqb = xq + (size_t)b * LQ * DD;
  for (int qc = 0; qc < 8; ++qc) {
    int q0 = qc * 32;
    int cs = (wv & 3) * 16, qsl = (wv >> 2) * 16;   // local subtiles
    int qg0 = q0 + qsl;
    v8f acc = {};
    for (int k0 = 0; k0 < DD; k0 += 4) {
      int kk = k0 + half * 2;
      v2f a = { xcl[(cs + lr) * XCSTR + kk] * w2l[kk],
                xcl[(cs + lr) * XCSTR + kk + 1] * w2l[kk + 1] };
      v2f bb = *(const v2f*)(xqb + (size_t)(qg0 + lr) * DD + kk);
      acc = wmma_f32(a, bb, acc);
    }
    int qg = qg0 + lr;
    float cmx = cmxl[qg], psum = 0.f;
    for (int r = 0; r < 8; ++r) {
      int cl = cs + half * 8 + r;
      float p = __expf(acc[r] + s0l[cl] + cml[cl] + s1l[qg] - cmx);
      pst[cl * PSTR + qsl + lr] = p;
      psum += p;
    }
    psum += __shfl_xor(psum, 16, 32);
    if (lane < 16) atomicAdd(&colsum[b * LQ + qg], psum);
    __syncthreads();
    // A(16q x 16d) += p^T (q x 64c) @ xc (64c x d): 16 tiles, 2 per wave
    for (int t = 0; t < 2; ++t) {
      int qs2 = (wv >> 2) * 16;
      int ds  = ((wv & 3) * 2 + t) * 16;
      v8f a2 = {};
      for (int k0 = 0; k0 < 64; k0 += 4) {
        int kk = k0 + half * 2;
        v2f a  = { pst[kk * PSTR + qs2 + lr], pst[(kk + 1) * PSTR + qs2 + lr] };
        v2f bb = { xcl[kk * XCSTR + ds + lr], xcl[(kk + 1) * XCSTR + ds + lr] };
        a2 = wmma_f32(a, bb, a2);
      }
      for (int r = 0; r < 8; ++r) {
        int qgA = q0 + qs2 + half * 8 + r;
        atomicAdd(&Aws[(size_t)(b * LQ + qgA) * DD + ds + lr], a2[r]);
      }
    }
    __syncthreads();
  }
}

// ---------------------------------------------------------------------------
// k3: row softmax over q + c2q / q2c GEMMs + write all 4 output slabs
// ---------------------------------------------------------------------------
__global__ __launch_bounds__(128) void k3_out(
    const float* __restrict__ xc, const float* __restrict__ xq,
    const int* __restrict__ qmask, const float* __restrict__ W2,
    const float* __restrict__ s0, const float* __restrict__ s1q,
    const float* __restrict__ colsum, const float* __restrict__ Aws,
    float* __restrict__ out) {
  __shared__ float xcl[32 * XCSTR];
  __shared__ float pl[32 * PSTR3];             // raw S then exp(S-max)
  __shared__ float w2l[DD], s0l[32], s1l[LQ], qml[LQ], cinv[LQ], rinv[32];
  int b = blockIdx.x >> 6, ct = blockIdx.x & 63, c0 = ct * 32;
  int tid = threadIdx.x;                        // 128 threads = 4 waves
  for (int i = tid; i < 32 * DD; i += 128) {
    int r = i >> 7, c = i & 127;
    xcl[r * XCSTR + c] = xc[(size_t)(b * LC + c0 + r) * DD + c];
  }
  if (tid < DD) w2l[tid] = W2[tid];
  if (tid < 32) s0l[tid] = s0[b * LC + c0 + tid];
  for (int i = tid; i < LQ; i += 128) {
    s1l[i]  = s1q[b * LQ + i];
    qml[i]  = qmask[b * LQ + i] ? 0.f : MASKV;
    cinv[i] = 1.0f / colsum[b * LQ + i];
  }
  __syncthreads();

  int lane = tid & 31, wv = tid >> 5, half = lane >> 4, lr = lane & 15;
  const float* xqb = xq + (size_t)b * LQ * DD;
  // Phase A: masked S into pl (32 tiles = 2c x 16q, 8 per wave)
  for (int t = 0; t < 8; ++t) {
    int id = wv * 8 + t;
    int cs = (id & 1) * 16;
    int qs = (id >> 1) * 16;
    v8f acc = {};
    for (int k0 = 0; k0 < DD; k0 += 4) {
      int kk = k0 + half * 2;
      v2f a = { xcl[(cs + lr) * XCSTR + kk] * w2l[kk],
                xcl[(cs + lr) * XCSTR + kk + 1] * w2l[kk + 1] };
      v2f bb = *(const v2f*)(xqb + (size_t)(qs + lr) * DD + kk);
      acc = wmma_f32(a, bb, acc);
    }
    int qg = qs + lr;
    for (int r = 0; r < 8; ++r) {
      int cl = cs + half * 8 + r;
      pl[cl * PSTR3 + qg] = acc[r] + s0l[cl] + s1l[qg] + qml[qg];
    }
  }
  __syncthreads();
  // Phase B: per-row softmax stats (wave 0; stride 258 -> conflict-free)
  if (tid < 32) {
    float m = -3.0e38f;
    for (int q = 0; q < LQ; ++q) m = fmaxf(m, pl[tid * PSTR3 + q]);
    float s = 0.f;
    for (int q = 0; q < LQ; ++q) {
      float e = __expf(pl[tid * PSTR3 + q] - m);
      pl[tid * PSTR3 + q] = e;
      s += e;
    }
    rinv[tid] = 1.0f / s;
  }
  __syncthreads();
  // Phase C1: c2q = P @ xq  (16 tiles = 2c x 8d, 4 per wave)
  for (int t = 0; t < 4; ++t) {
    int id = wv * 4 + t;
    int cs = (id & 1) * 16;
    int ds = (id >> 1) * 16;
    v8f acc = {};
    for (int k0 = 0; k0 < LQ; k0 += 4) {
      int kk = k0 + half * 2;
      v2f a  = *(const v2f*)&pl[(cs + lr) * PSTR3 + kk];
      v2f bb = { xqb[(size_t)kk * DD + ds + lr],
                 xqb[(size_t)(kk + 1) * DD + ds + lr] };
      acc = wmma_f32(a, bb, acc);
    }
    for (int r = 0; r < 8; ++r) {
      int cl  = cs + half * 8 + r;
      float v = acc[r] * rinv[cl];
      float x = xcl[cl * XCSTR + ds + lr];
      size_t o = (size_t)(b * LC + c0 + cl) * 512;
      out[o + 128 + ds + lr] = v;        // c2q
      out[o + 256 + ds + lr] = x * v;    // xc * c2q
    }
  }
  // Phase C2: q2c = P @ (A/colsum)
  const float* Ab = Aws + (size_t)b * LQ * DD;
  for (int t = 0; t < 4; ++t) {
    int id = wv * 4 + t;
    int cs = (id & 1) * 16;
    int ds = (id >> 1) * 16;
    v8f acc = {};
    for (int k0 = 0; k0 < LQ; k0 += 4) {
      int kk = k0 + half * 2;
      v2f a  = *(const v2f*)&pl[(cs + lr) * PSTR3 + kk];
      v2f bb = { Ab[(size_t)kk * DD + ds + lr] * cinv[kk],
                 Ab[(size_t)(kk + 1) * DD + ds + lr] * cinv[kk + 1] };
      acc = wmma_f32(a, bb, acc);
    }
    for (int r = 0; r < 8; ++r) {
      int cl  = cs + half * 8 + r;
      float v = acc[r] * rinv[cl];
      float x = xcl[cl * XCSTR + ds + lr];
      size_t o = (size_t)(b * LC + c0 + cl) * 512;
      out[o + 384 + ds + lr] = x * v;    // xc * q2c
    }
  }
  // x_cont passthrough (cols 0..127)
  for (int i = tid; i < 32 * DD; i += 128) {
    int r = i >> 7, c = i & 127;
    out[(size_t)(b * LC + c0 + r) * 512 + c] = xcl[r * XCSTR + c];
  }
}

// ---------------------------------------------------------------------------
extern "C" void kernel_launch(void* const* d_in, const int* in_sizes, int n_in,
                              void* d_out, int out_size, void* d_ws, size_t ws_size,
                              hipStream_t stream) {
  const float* xc    = (const float*)d_in[0];
  const float* xq    = (const float*)d_in[1];
  const int*   cmask = (const int*)d_in[2];
  const int*   qmask = (const int*)d_in[3];
  const float* W0    = (const float*)d_in[4];
  const float* W1    = (const float*)d_in[5];
  const float* W2    = (const float*)d_in[6];
  const float* bias  = (const float*)d_in[7];
  float* out = (float*)d_out;

  float* s0   = (float*)d_ws;            // [B*Lc]
  float* s1q  = s0   + B_ * LC;          // [B*Lq]  (bias folded in)
  float* cmx  = s1q  + B_ * LQ;          // [B*Lq]
  float* csum = cmx  + B_ * LQ;          // [B*Lq]
  float* Aws  = csum + B_ * LQ;          // [B*Lq*D]

  k0_init  <<<(B_ * LQ * DD + 255) / 256, 256, 0, stream>>>(
      xc, xq, W0, W1, bias, s0, s1q, cmx, csum, Aws);
  k1_colmax<<<B_ * (LC / 64), 256, 0, stream>>>(
      xc, xq, cmask, W2, s0, s1q, cmx);
  k2_colacc<<<B_ * (LC / 64), 256, 0, stream>>>(
      xc, xq, cmask, W2, s0, s1q, cmx, csum, Aws);
  k3_out   <<<B_ * (LC / 32), 128, 0, stream>>>(
      xc, xq, qmask, W2, s0, s1q, csum, Aws, out);
}